// WLS_36799279792921
// MI455X (gfx1250) — compile-verified
//
#include <hip/hip_runtime.h>

typedef __attribute__((ext_vector_type(2))) float v2f;
typedef __attribute__((ext_vector_type(8))) float v8f;

#define WLS_EPS 1e-6f

// Cross-half swap: every lane gets the partner value from lane^16.
// Raw v_permlanex16_b32 (identity nibble table) avoids the builtin's tied
// 'old' operand, which was costing a v_mov per swap. EXEC is all-ones here.
__device__ __forceinline__ float wls_xswap(float v) {
  float r;
  asm("v_permlanex16_b32 %0, %1, %2, %3"
      : "=v"(r)
      : "v"(v), "s"(0x76543210), "s"(0xfedcba98u));
  return r;
}

// Single-instruction max(v, 0): v_max_num_f32 returns the number when one
// input is NaN, matching fmaxf(v, 0) without the canonicalize the compiler
// otherwise inserts (fmaxf was lowering to 2 VALU ops).
__device__ __forceinline__ float wls_relu(float v) {
  float r;
  asm("v_max_num_f32 %0, 0, %1" : "=v"(r) : "v"(v));
  return r;
}

// GEMM2 on the matrix pipe: acc += W2^T_pad @ relu(h^T).
// hr[] = relu'd GEMM1 D (C/D layout: VGPR v = channel v+8*hi, lane = row).
// B layout (4x16 chunk): V0 = K{4c lo / 4c+2 hi}, V1 = K{4c+1 lo / 4c+3 hi};
// channel k<8 lives in low halves of hr[k], k>=8 in high halves of hr[k-8].
__device__ __forceinline__ v8f wls_gemm2(const v2f* a2, const float* hr,
                                         int hi, v8f acc) {
  float sh[8];
#pragma unroll
  for (int v = 0; v < 8; ++v) sh[v] = wls_xswap(hr[v]);
  v2f b;
  b.x = hi ? sh[2] : hr[0]; b.y = hi ? sh[3] : hr[1];   // K = 0..3
  acc = __builtin_amdgcn_wmma_f32_16x16x4_f32(false, a2[0], false, b, (short)0, acc, false, false);
  b.x = hi ? sh[6] : hr[4]; b.y = hi ? sh[7] : hr[5];   // K = 4..7
  acc = __builtin_amdgcn_wmma_f32_16x16x4_f32(false, a2[1], false, b, (short)0, acc, false, false);
  b.x = hi ? hr[2] : sh[0]; b.y = hi ? hr[3] : sh[1];   // K = 8..11
  acc = __builtin_amdgcn_wmma_f32_16x16x4_f32(false, a2[2], false, b, (short)0, acc, false, false);
  b.x = hi ? hr[6] : sh[4]; b.y = hi ? hr[7] : sh[5];   // K = 12..15
  acc = __builtin_amdgcn_wmma_f32_16x16x4_f32(false, a2[3], false, b, (short)0, acc, false, false);
  return acc;
}

// One pair = two 16-row tiles (32 rows). Low lanes finalize tile A's rows,
// high lanes tile B's rows. CLAMPED=true only for the (rare) partial last pair.
template <bool CLAMPED>
static __device__ __forceinline__ void wls_pair(
    const float* __restrict__ x, float* __restrict__ out,
    v2f a0, v2f a1, const v2f* a2, v8f cb1, v8f cb2,
    int p, int hi, int lo, int Brows)
{
  const int rowA  = p * 32 + lo;
  const int rowB  = rowA + 16;
  const int rowme = p * 32 + hi * 16 + lo;

  uint32_t oA0, oB0, orow;
  if (CLAMPED) {
    const int rcA = rowA  < Brows ? rowA  : Brows - 1;   // keep EXEC full for WMMA
    const int rcB = rowB  < Brows ? rowB  : Brows - 1;
    const int rcm = rowme < Brows ? rowme : Brows - 1;
    oA0  = (uint32_t)rcA * 8 + 2 * hi;
    oB0  = (uint32_t)rcB * 8 + 2 * hi;
    orow = (uint32_t)rcm * 8;
  } else {
    oA0  = (uint32_t)rowA * 8 + 2 * hi;
    oB0  = oA0 + 128;
    orow = (uint32_t)rowme * 8;
  }

  // B = x^T chunks (lane = contiguous float2 x[row][4c+2hi..+1]) + this lane's
  // full x row for the solver (cache-hot reload instead of 20 shuffle/selects)
  v2f bxA0 = *(const v2f*)(x + oA0);
  v2f bxA1 = *(const v2f*)(x + oA0 + 4);
  v2f bxB0 = *(const v2f*)(x + oB0);
  v2f bxB1 = *(const v2f*)(x + oB0 + 4);
  float4 xr0 = *(const float4*)(x + orow);
  float4 xr1 = *(const float4*)(x + orow + 4);

  // GEMM1: h^T = W1^T @ x^T + b1 (exact fp32 on matrix pipe)
  v8f hA = cb1, hB = cb1;
  hA = __builtin_amdgcn_wmma_f32_16x16x4_f32(false, a0, false, bxA0, (short)0, hA, false, false);
  hA = __builtin_amdgcn_wmma_f32_16x16x4_f32(false, a1, false, bxA1, (short)0, hA, false, false);
  hB = __builtin_amdgcn_wmma_f32_16x16x4_f32(false, a0, false, bxB0, (short)0, hB, false, false);
  hB = __builtin_amdgcn_wmma_f32_16x16x4_f32(false, a1, false, bxB1, (short)0, hB, false, false);

  float hrA[8], hrB[8];
#pragma unroll
  for (int v = 0; v < 8; ++v) {
    hrA[v] = wls_relu(hA[v]);
    hrB[v] = wls_relu(hB[v]);
  }

  // GEMM2: sigma^T; low lane n holds all 8 channels of its tile's row n
  v8f sA = wls_gemm2(a2, hrA, hi, cb2);
  v8f sB = wls_gemm2(a2, hrB, hi, cb2);

  // merge tiles (low lanes: tile A row, high lanes: tile B row) + softplus
  float sg[8];
#pragma unroll
  for (int v = 0; v < 8; ++v) {
    float zb = wls_xswap(sB[v]);
    float z  = hi ? zb : sA[v];
    sg[v] = wls_relu(z) + __logf(1.0f + __expf(-fabsf(z))) + WLS_EPS;
  }

  const float xf[8] = {xr0.x, xr0.y, xr0.z, xr0.w, xr1.x, xr1.y, xr1.z, xr1.w};

  // Sherman-Morrison: StS = diag(d1..d7) + d0*11^T ; 1/d_j = sg_j
  float d0 = 1.0f / sg[0];
  float a  = d0 * xf[0];
  float q = 0.0f, sumx = 0.0f;
#pragma unroll
  for (int j = 1; j < 8; ++j) { q += sg[j]; sumx += xf[j]; }
  float s  = a * q + sumx;
  float g  = d0 * s / (1.0f + d0 * q);
  float o_first = s - g * q;
  float beta[7];
#pragma unroll
  for (int j = 0; j < 7; ++j) beta[j] = (a - g) * sg[j + 1] + xf[j + 1];

  // each lane stores one full output row (two b128s, contiguous per half-wave)
  if (!CLAMPED || rowme < Brows) {
    float* op = out + orow;
    *(float4*)(op)     = make_float4(o_first, beta[0], beta[1], beta[2]);
    *(float4*)(op + 4) = make_float4(beta[3], beta[4], beta[5], beta[6]);
  }
}

__global__ __launch_bounds__(256) void wls_wmma_kernel(
    const float* __restrict__ x, const float* __restrict__ W1,
    const float* __restrict__ b1, const float* __restrict__ W2,
    const float* __restrict__ b2, float* __restrict__ out, int Brows)
{
  const int lane = threadIdx.x & 31;
  const int wave = threadIdx.x >> 5;
  const int hi   = lane >> 4;
  const int lo   = lane & 15;

  const int wavesPerBlock = blockDim.x >> 5;
  const int gwave  = blockIdx.x * wavesPerBlock + wave;
  const int nwaves = gridDim.x * wavesPerBlock;
  const int nfull  = Brows >> 5;          // full 32-row pairs
  const int rem    = Brows & 31;

  // GEMM1 A = W1^T (16x8): A_c[m][k] = W1[4c+k][m]; lane holds (K=2hi, 2hi+1)
  v2f a0, a1;
  a0.x = W1[(0 + 2 * hi) * 16 + lo];
  a0.y = W1[(1 + 2 * hi) * 16 + lo];
  a1.x = W1[(4 + 2 * hi) * 16 + lo];
  a1.y = W1[(5 + 2 * hi) * 16 + lo];

  // GEMM2 A = W2^T (8x16) zero-padded to 16 rows: A2_c[m][k] = m<8 ? W2[4c+k][m] : 0
  v2f a2[4];
#pragma unroll
  for (int c = 0; c < 4; ++c) {
    a2[c].x = (lo < 8) ? W2[(4 * c + 2 * hi    ) * 8 + lo] : 0.0f;
    a2[c].y = (lo < 8) ? W2[(4 * c + 2 * hi + 1) * 8 + lo] : 0.0f;
  }

  // C inits: h^T row m gets b1[m] (m = v + 8hi); sigma^T row j gets b2[j]
  v8f cb1, cb2;
#pragma unroll
  for (int v = 0; v < 8; ++v) { cb1[v] = b1[8 * hi + v]; cb2[v] = b2[v]; }

  for (int p = gwave; p < nfull; p += nwaves) {
    if (p + nwaves < nfull)  // prefetch next pair's rows (global_prefetch_b8)
      __builtin_prefetch(x + (size_t)((p + nwaves) * 32 + lane) * 8, 0, 3);
    wls_pair<false>(x, out, a0, a1, a2, cb1, cb2, p, hi, lo, Brows);
  }
  // partial last pair (only if Brows % 32 != 0), done by the wave whose
  // strided sequence owns pair index nfull
  if (rem != 0 && gwave == (nfull % nwaves))
    wls_pair<true>(x, out, a0, a1, a2, cb1, cb2, nfull, hi, lo, Brows);
}

extern "C" void kernel_launch(void* const* d_in, const int* in_sizes, int n_in,
                              void* d_out, int out_size, void* d_ws, size_t ws_size,
                              hipStream_t stream) {
  const float* x  = (const float*)d_in[0];
  const float* W1 = (const float*)d_in[1];
  const float* b1 = (const float*)d_in[2];
  const float* W2 = (const float*)d_in[3];
  const float* b2 = (const float*)d_in[4];
  float* out = (float*)d_out;

  const int Brows = in_sizes[0] / 8;          // 524288
  const int npTot = (Brows + 31) / 32;        // 32 rows per wave-iteration
  // 8 waves/block, ~4 pairs per wave via grid-stride
  int nblocks = (npTot + 31) / 32;
  if (nblocks < 1) nblocks = 1;
  if (nblocks > 32768) nblocks = 32768;

  wls_wmma_kernel<<<nblocks, 256, 0, stream>>>(x, W1, b1, W2, b2, out, Brows);
}